// EncoderSpatial_75814762709662
// MI455X (gfx1250) — compile-verified
//
#include <hip/hip_runtime.h>

typedef __attribute__((ext_vector_type(2))) float v2f;
typedef __attribute__((ext_vector_type(8))) float v8f;

#define B_ 32
#define N_ 64
#define T_ 50
#define F_ 2
#define E_ 8
#define NGRAPH (B_ * T_)   // 1600 graphs
#define WAVES 8            // one wave32 per graph, 8 graphs per block

// w = min(1/dist, 1) with dist clamped at 1e-6  ==  min(rsqrt(max(d2,1e-12)), 1)
// v_rsq_f32 is a single TRANS op (co-executes with VALU on CDNA5).
__device__ __forceinline__ float edge_weight(float d2) {
    return fminf(__builtin_amdgcn_rsqf(fmaxf(d2, 1e-12f)), 1.0f);
}

__global__ __launch_bounds__(WAVES * 32)
void enc_spatial_kernel(const float* __restrict__ x,     // (B,N,T,F)
                        const float* __restrict__ Wres,  // (F,E)
                        const float* __restrict__ Wgcn,  // (F,E)
                        const float* __restrict__ bgcn,  // (E,)
                        float* __restrict__ out)         // [3200 inputs][12800 enc]
{
    __shared__ float s_p[WAVES][N_][2];
    __shared__ float s_nrm[WAVES][N_];
    __shared__ float s_deg[WAVES][N_];
    __shared__ float s_dinv[WAVES][N_];

    const int wave = threadIdx.x >> 5;
    const int lane = threadIdx.x & 31;
    const int g = blockIdx.x * WAVES + wave;   // graph id = b*T + t
    const int b = g / T_;
    const int t = g % T_;

    // zero per-column degree accumulators (64 per wave)
    s_deg[wave][lane]      = 0.0f;
    s_deg[wave][lane + 32] = 0.0f;

    // Build WMMA fragments. Tile tt covers nodes 16*tt .. 16*tt+15.
    // A(16x4 f32) layout: lanes 0-15 hold K=0,1 (the real x,y coords) for row M=lane;
    // lanes 16-31 hold K=2,3 which we zero-pad. B(4x16) has identical lane content,
    // so one fragment per tile serves both operands of G = P * P^T.
    v2f frag[4];
#pragma unroll
    for (int tt = 0; tt < 4; ++tt) {
        v2f v = {0.0f, 0.0f};
        if (lane < 16) {
            int n = tt * 16 + lane;
            const float* p = x + (((size_t)b * N_ + n) * T_ + t) * F_;
            v.x = p[0];
            v.y = p[1];
            s_p[wave][n][0] = v.x;
            s_p[wave][n][1] = v.y;
            s_nrm[wave][n]  = v.x * v.x + v.y * v.y;
        }
        frag[tt] = v;
    }
    __syncthreads();

    // 16 Gram tiles per graph via V_WMMA_F32_16X16X4_F32 (K padded 2 -> 4).
    // dist^2(i,j) = |p_i|^2 + |p_j|^2 - 2*G[i][j]
    v8f czero = {};
#pragma unroll
    for (int r = 0; r < 4; ++r) {
#pragma unroll
        for (int c = 0; c < 4; ++c) {
            v8f d = __builtin_amdgcn_wmma_f32_16x16x4_f32(
                false, frag[r], false, frag[c], (short)0, czero, false, false);
            // D layout: col j = 16c + (lane&15); rows 16r + v (+8 for lanes>=16)
            const int j     = 16 * c + (lane & 15);
            const int rbase = 16 * r + ((lane >> 4) << 3);
            const float nj  = s_nrm[wave][j];
            float colsum = 0.0f;
#pragma unroll
            for (int v = 0; v < 8; ++v) {
                const int i  = rbase + v;
                float d2     = s_nrm[wave][i] + nj - 2.0f * d[v];
                float w      = edge_weight(d2);
                colsum      += (i == j) ? 0.0f : w;   // exclude self edge
            }
            atomicAdd(&s_deg[wave][j], colsum);       // ds_add_f32
        }
    }
    __syncthreads();

    // dinv = (deg + 1)^-0.5  -> single v_rsq_f32
    s_dinv[wave][lane]      = __builtin_amdgcn_rsqf(s_deg[wave][lane] + 1.0f);
    s_dinv[wave][lane + 32] = __builtin_amdgcn_rsqf(s_deg[wave][lane + 32] + 1.0f);
    __syncthreads();

    // agg for node 0: s = sum_i c_i * p_i with
    //   c_0 = dinv0^2,  c_i = dinv[i]*w(i,0)*dinv0  (i != 0)
    const float p0x   = s_p[wave][0][0];
    const float p0y   = s_p[wave][0][1];
    const float dinv0 = s_dinv[wave][0];
    float sx = 0.0f, sy = 0.0f;
#pragma unroll
    for (int k = 0; k < 2; ++k) {
        const int i = lane + 32 * k;
        const float px = s_p[wave][i][0];
        const float py = s_p[wave][i][1];
        float ci;
        if (i == 0) {
            ci = dinv0 * dinv0;
        } else {
            const float dx = px - p0x, dy = py - p0y;
            const float w = edge_weight(dx * dx + dy * dy);
            ci = s_dinv[wave][i] * w * dinv0;
        }
        sx += ci * px;
        sy += ci * py;
    }
    // wave32 reduction
#pragma unroll
    for (int off = 16; off > 0; off >>= 1) {
        sx += __shfl_down(sx, off, 32);
        sy += __shfl_down(sy, off, 32);
    }

    if (lane == 0) {
        // output 0: inputs = x[:,0] -> (B,T,F); node 0 of graph == x[b,0,t,:]
        out[g * 2 + 0] = p0x;
        out[g * 2 + 1] = p0y;
        // output 1: enc_sp[b,t,:] = relu(s@W_gcn + b_gcn + p0@W_res)
        float* enc = out + B_ * T_ * F_ + (size_t)g * E_;
#pragma unroll
        for (int e = 0; e < E_; ++e) {
            float val = sx * Wgcn[e] + sy * Wgcn[E_ + e] + bgcn[e]
                      + p0x * Wres[e] + p0y * Wres[E_ + e];
            enc[e] = fmaxf(val, 0.0f);
        }
    }
}

extern "C" void kernel_launch(void* const* d_in, const int* in_sizes, int n_in,
                              void* d_out, int out_size, void* d_ws, size_t ws_size,
                              hipStream_t stream) {
    const float* x    = (const float*)d_in[0];
    const float* Wres = (const float*)d_in[1];
    const float* Wgcn = (const float*)d_in[2];
    const float* bgcn = (const float*)d_in[3];
    float* out        = (float*)d_out;

    dim3 grid(NGRAPH / WAVES);   // 200 blocks, exact fit (no divergent barriers)
    dim3 block(WAVES * 32);      // 256 threads = 8 wave32
    hipLaunchKernelGGL(enc_spatial_kernel, grid, block, 0, stream,
                       x, Wres, Wgcn, bgcn, out);
}